// GraphConvolutionLayer_22428319219855
// MI455X (gfx1250) — compile-verified
//
#include <hip/hip_runtime.h>
#include <math.h>

typedef __attribute__((ext_vector_type(2))) float v2f;
typedef __attribute__((ext_vector_type(8))) float v8f;

#define K_DIM 512
#define N_DIM 256
#define KB    32
#define MT    64
#define NT    64
#define NBLK  (K_DIM / KB)

// LDS byte-offset of a pointer into a __shared__ array (generic -> AS3 cast)
__device__ __forceinline__ unsigned lds_off(const void* p) {
    return (unsigned)(unsigned long long)(const __attribute__((address_space(3))) void*)p;
}

// ---------------------------------------------------------------- zero output
__global__ void zero_f4_kernel(float4* __restrict__ p, int n4) {
    int i = blockIdx.x * blockDim.x + threadIdx.x;
    int stride = gridDim.x * blockDim.x;
    float4 z = make_float4(0.f, 0.f, 0.f, 0.f);
    for (; i < n4; i += stride) p[i] = z;
}

// --------------------------------------------- GEMM (f32 WMMA) + tanh fusion
// support[row][col] = tanh( sum_k x[row][k] * w[k][col] )
// Double-buffered async global->LDS staging (ASYNCcnt), WMMA f32 16x16x4 math.
__global__ __launch_bounds__(256)
void gemm_tanh_wmma_kernel(const float* __restrict__ x,
                           const float* __restrict__ w,
                           float* __restrict__ support,
                           int nrows,
                           const int* __restrict__ act_p) {
    __shared__ __align__(16) float Als[2][MT][KB + 4];   // 2 x 64 x 36
    __shared__ __align__(16) float Bls[2][KB][NT + 4];   // 2 x 32 x 68

    const int tid   = threadIdx.x;
    const int lane  = tid & 31;
    const int wave  = tid >> 5;        // 0..7
    const int waveM = wave >> 1;       // 0..3 : 16-row band
    const int waveN = wave & 1;        // 0..1 : 32-col band

    const int rowBase = blockIdx.x * MT;
    const int colBase = blockIdx.y * NT;
    const int act     = *act_p;

    v8f acc0 = {};
    v8f acc1 = {};

    // cooperative tile-load assignments (each thread DMAs 2x16B per tile)
    const int arow = tid >> 2;         // 0..63   (4 threads per A row)
    const int acol = (tid & 3) * 8;    // 0,8,16,24
    const int brow = tid >> 3;         // 0..31   (8 threads per B row)
    const int bcol = (tid & 7) * 8;    // 0..56

    // clamp tail rows: garbage A rows only feed C rows that are never stored
    int gr = rowBase + arow;
    if (gr >= nrows) gr = nrows - 1;
    const unsigned ga_row = (unsigned)(((size_t)gr * K_DIM + acol) * sizeof(float));
    const unsigned gb_row = (unsigned)(((size_t)brow * N_DIM + colBase + bcol) * sizeof(float));

    const int mlane = lane & 15;
    const int khalf = (lane >> 4) * 2; // lanes 0-15 -> K 0,1 ; lanes 16-31 -> K 2,3

    // issue one K-slab's async DMA: exactly 4 wave-level async instructions
    auto issue_stage = [&](int buf, int k0) {
        unsigned ga = ga_row + (unsigned)(k0 * sizeof(float));
        unsigned la = lds_off(&Als[buf][arow][acol]);
        asm volatile("global_load_async_to_lds_b128 %0, %1, %2 offset:0"
                     :: "v"(la), "v"(ga), "s"(x) : "memory");
        asm volatile("global_load_async_to_lds_b128 %0, %1, %2 offset:16"
                     :: "v"(la), "v"(ga), "s"(x) : "memory");
        unsigned gb = gb_row + (unsigned)(k0 * N_DIM * sizeof(float));
        unsigned lb = lds_off(&Bls[buf][brow][bcol]);
        asm volatile("global_load_async_to_lds_b128 %0, %1, %2 offset:0"
                     :: "v"(lb), "v"(gb), "s"(w) : "memory");
        asm volatile("global_load_async_to_lds_b128 %0, %1, %2 offset:16"
                     :: "v"(lb), "v"(gb), "s"(w) : "memory");
    };

    issue_stage(0, 0);

    for (int blk = 0; blk < NBLK; ++blk) {
        const int nxt = blk + 1;
        if (nxt < NBLK) {
            issue_stage(nxt & 1, nxt * KB);
            // previous stage (4 asyncs) complete when <=4 remain outstanding
            asm volatile("s_wait_asynccnt 0x4" ::: "memory");
        } else {
            asm volatile("s_wait_asynccnt 0x0" ::: "memory");
        }
        __syncthreads();

        const float (*As)[KB + 4] = Als[blk & 1];
        const float (*Bs)[NT + 4] = Bls[blk & 1];

        #pragma unroll
        for (int kk = 0; kk < KB; kk += 4) {
            const int kb = kk + khalf;
            v2f a;
            a.x = As[waveM * 16 + mlane][kb];
            a.y = As[waveM * 16 + mlane][kb + 1];
            v2f b0;
            b0.x = Bs[kb][waveN * 32 + mlane];
            b0.y = Bs[kb + 1][waveN * 32 + mlane];
            v2f b1;
            b1.x = Bs[kb][waveN * 32 + 16 + mlane];
            b1.y = Bs[kb + 1][waveN * 32 + 16 + mlane];
            acc0 = __builtin_amdgcn_wmma_f32_16x16x4_f32(
                false, a, false, b0, (short)0, acc0, false, false);
            acc1 = __builtin_amdgcn_wmma_f32_16x16x4_f32(
                false, a, false, b1, (short)0, acc1, false, false);
        }
        __syncthreads();
    }

    // ---- epilogue: tanh + store (C/D layout: M = v + 8*(lane>=16), N = lane%16)
    const int rhalf = (lane >> 4) * 8;
    #pragma unroll
    for (int v = 0; v < 8; v++) {
        int row = rowBase + waveM * 16 + rhalf + v;
        if (row < nrows) {
            float s0 = acc0[v];
            float s1 = acc1[v];
            if (act) { s0 = tanhf(s0); s1 = tanhf(s1); }
            size_t base = (size_t)row * N_DIM + colBase + waveN * 32;
            support[base + mlane]      = s0;
            support[base + 16 + mlane] = s1;
        }
    }
}

// ------------------------------------------------- SpMM edge scatter (atomics)
// out[r] += vals[e] * support[c], one wave32 per edge, 8 floats per lane.
// Edge index forced wave-uniform so edge metadata becomes scalar (SMEM) loads.
__global__ __launch_bounds__(256)
void spmm_scatter_kernel(const int* __restrict__ erow,
                         const int* __restrict__ ecol,
                         const float* __restrict__ eval,
                         const float* __restrict__ support,
                         float* __restrict__ out,
                         int E) {
    const int lane = threadIdx.x & 31;
    const int wave = __builtin_amdgcn_readfirstlane(
        blockIdx.x * (blockDim.x >> 5) + (threadIdx.x >> 5));
    const int nwaves = gridDim.x * (blockDim.x >> 5);

    for (int e = wave; e < E; e += nwaves) {
        const int   r = erow[e];
        const int   c = ecol[e];
        const float v = eval[e];

        const float4* src =
            reinterpret_cast<const float4*>(support + (size_t)c * N_DIM) + lane * 2;
        float4 s0 = src[0];
        float4 s1 = src[1];

        float* dst = out + (size_t)r * N_DIM + lane * 8;
        atomicAdd(dst + 0, s0.x * v);
        atomicAdd(dst + 1, s0.y * v);
        atomicAdd(dst + 2, s0.z * v);
        atomicAdd(dst + 3, s0.w * v);
        atomicAdd(dst + 4, s1.x * v);
        atomicAdd(dst + 5, s1.y * v);
        atomicAdd(dst + 6, s1.z * v);
        atomicAdd(dst + 7, s1.w * v);
    }
}

// ---------------------------------------------------------------------- launch
extern "C" void kernel_launch(void* const* d_in, const int* in_sizes, int n_in,
                              void* d_out, int out_size, void* d_ws, size_t ws_size,
                              hipStream_t stream) {
    const float* x    = (const float*)d_in[0];
    const float* w    = (const float*)d_in[1];
    const int*   erow = (const int*)d_in[2];
    const int*   ecol = (const int*)d_in[3];
    const float* eval = (const float*)d_in[4];
    const int*   act  = (const int*)d_in[5];

    float* out     = (float*)d_out;
    float* support = (float*)d_ws;   // [nrows, 256] f32 scratch (~102 MB)

    const int nrows = in_sizes[0] / K_DIM;
    const int E     = in_sizes[2];

    // 1) zero the accumulation target (harness poisons it)
    {
        int n4 = out_size / 4;
        int blocks = (n4 + 255) / 256;
        if (blocks > 4096) blocks = 4096;
        zero_f4_kernel<<<blocks, 256, 0, stream>>>((float4*)out, n4);
    }

    // 2) fused GEMM + tanh into workspace via f32 WMMA + async LDS staging
    {
        dim3 grid((nrows + MT - 1) / MT, N_DIM / NT);
        gemm_tanh_wmma_kernel<<<grid, 256, 0, stream>>>(x, w, support, nrows, act);
    }

    // 3) edge scatter-add (L2-resident gather + f32 atomics)
    {
        spmm_scatter_kernel<<<4096, 256, 0, stream>>>(erow, ecol, eval,
                                                      support, out, E);
    }
}